// CommunityTimeModel_61804579389717
// MI455X (gfx1250) — compile-verified
//
#include <hip/hip_runtime.h>
#include <math.h>

// ---------------------------------------------------------------------------
// CommunityTimeModel forward, collapsed to rank-1 structure:
//   Phase 1: per-edge scatter  -> acc[n] = (A_intra, B_intra, A_inter, B_inter)
//   Phase 2: [N,4] x [4,256] GEMM via V_WMMA_F32_16X16X4_F32 + SiLU combine
// Memory-bound (edge phase L2-resident; 50MB fp32 output dominates).
// ---------------------------------------------------------------------------

typedef __attribute__((ext_vector_type(2))) float v2f;
typedef __attribute__((ext_vector_type(8))) float v8f;

#define HIDDEN 64

__global__ void ctm_zero_kernel(float* __restrict__ acc, int n) {
    int i = blockIdx.x * blockDim.x + threadIdx.x;
    if (i < n) acc[i] = 0.0f;
}

__global__ void ctm_edge_scatter_kernel(const float* __restrict__ xr,
                                        const float* __restrict__ xi,
                                        const int*   __restrict__ src,
                                        const int*   __restrict__ dst,
                                        const float* __restrict__ ew,
                                        const int*   __restrict__ comm,
                                        float* __restrict__ acc, int E) {
    int e = blockIdx.x * blockDim.x + threadIdx.x;
    if (e >= E) return;
    int s = src[e];
    int d = dst[e];
    float w = ew[e];
    // intra-community edges hit slots {0,1}, inter-community slots {2,3}
    int off = (comm[s] == comm[d]) ? 0 : 2;
    float mr = w * xr[s];
    float mi = w * xi[s];
    float* base = acc + (size_t)d * 4 + off;
    atomicAdd(base + 0, mr);   // A accumulator
    atomicAdd(base + 1, mi);   // B accumulator
}

// SiLU with hardware reciprocal: one v_exp_f32 + one v_rcp_f32 + mul/add.
// (Plain 1.0f/x lowers to the precise div_scale/fma-correction sequence,
//  ~6 VALU ops/element — pointless bloat for a memory-bound kernel.)
__device__ __forceinline__ float silu_f(float x) {
    return x * __builtin_amdgcn_rcpf(1.0f + __expf(-x));
}

// One wave handles 16 nodes. A = S[16x4] (K=4 exact). For each of 8 column
// tiles (8 h-values x {real,imag}) do two WMMAs: B1 carries local weights in
// K-rows {0,1}, B2 carries global weights in K-rows {2,3}. Then
// out = silu(D1) + silu(D2) elementwise, stored into the [N,64,2] layout.
__global__ void ctm_expand_wmma_kernel(const float* __restrict__ acc,
                                       const float* __restrict__ Wlr,
                                       const float* __restrict__ Wli,
                                       const float* __restrict__ Wgr,
                                       const float* __restrict__ Wgi,
                                       float* __restrict__ out) {
    __shared__ float sW[4][HIDDEN];
    int tid = threadIdx.x;
    {
        int which = tid >> 6;      // 0..3  (256 threads cover 4*64 exactly)
        int h     = tid & 63;
        const float* W = (which == 0) ? Wlr : (which == 1) ? Wli
                       : (which == 2) ? Wgr : Wgi;
        sW[which][h] = W[h];
    }
    __syncthreads();

    const int wave  = tid >> 5;
    const int lane  = tid & 31;
    const int chunk = blockIdx.x * (blockDim.x >> 5) + wave;
    const int node0 = chunk * 16;

    const int m     = lane & 15;   // node row within chunk (A) / column (B,D)
    const int khalf = lane >> 4;   // 0: K={0,1}, 1: K={2,3}
    const int kbase = khalf * 2;

    // A-matrix 16x4 F32 operand per ISA layout:
    //   lanes 0-15: VGPR0=K0, VGPR1=K1 ; lanes 16-31: VGPR0=K2, VGPR1=K3
    const float* ap = acc + ((size_t)(node0 + m)) * 4 + kbase;
    v2f a;
    a.x = ap[0];
    a.y = ap[1];

    const int col = lane & 15;     // column within 16-wide tile
    const int hl  = col >> 1;      // local h (0..7)
    const int p   = col & 1;       // 0 = real part, 1 = imag part
    const float fkh = (khalf == 0) ? 1.0f : 0.0f;   // branchless masks
    const float fkg = 1.0f - fkh;

    for (int t = 0; t < 8; ++t) {
        int h = t * 8 + hl;
        float wlr = sW[0][h], wli = sW[1][h];
        float wgr = sW[2][h], wgi = sW[3][h];

        // B-matrix 4x16 F32: lanes hold N, VGPR pair holds (Kbase, Kbase+1).
        // Local GNN:  ar = A*Wlr - B*Wli ; ai = A*Wli + B*Wlr
        float k0 = p ? wli : wlr;
        float k1 = p ? wlr : -wli;
        // Global GNN: ar = A*Wgr - B*Wgi ; ai = A*Wgi + B*Wgr
        float k2 = p ? wgi : wgr;
        float k3 = p ? wgr : -wgi;

        v2f b1, b2;
        b1.x = fkh * k0;  b1.y = fkh * k1;   // zero on K rows 2,3
        b2.x = fkg * k2;  b2.y = fkg * k3;   // zero on K rows 0,1

        v8f c0 = {};
        v8f d1 = __builtin_amdgcn_wmma_f32_16x16x4_f32(
            false, a, false, b1, (short)0, c0, false, false);
        v8f d2 = __builtin_amdgcn_wmma_f32_16x16x4_f32(
            false, a, false, b2, (short)0, c0, false, false);

        // D 16x16 f32: VGPR v -> row (v + 8*khalf), column = lane&15.
        // Output offset within node row = h*2 + p = t*16 + col (contiguous!).
        int colOut = t * 16 + col;
        #pragma unroll
        for (int v = 0; v < 8; ++v) {
            int row = v + 8 * khalf;
            float r = silu_f(d1[v]) + silu_f(d2[v]);
            out[((size_t)(node0 + row)) * (HIDDEN * 2) + colOut] = r;
        }
    }
}

extern "C" void kernel_launch(void* const* d_in, const int* in_sizes, int n_in,
                              void* d_out, int out_size, void* d_ws, size_t ws_size,
                              hipStream_t stream) {
    const float* xr   = (const float*)d_in[0];
    const float* xi   = (const float*)d_in[1];
    const int*   ei   = (const int*)  d_in[2];   // [2, E]
    const float* ew   = (const float*)d_in[3];
    const int*   comm = (const int*)  d_in[4];
    const float* Wlr  = (const float*)d_in[5];
    const float* Wli  = (const float*)d_in[6];
    const float* Wgr  = (const float*)d_in[7];
    const float* Wgi  = (const float*)d_in[8];
    float* out = (float*)d_out;

    const int N = in_sizes[0];   // 98304 (x_real is [N,1])
    const int E = in_sizes[3];   // 1572864 (edge_weight is [E])

    float* acc = (float*)d_ws;   // [N][4] fp32 accumulators (1.5 MB)

    // Phase 0: zero accumulators (graph-capture safe: plain kernel on stream)
    int accN = 4 * N;
    ctm_zero_kernel<<<(accN + 255) / 256, 256, 0, stream>>>(acc, accN);

    // Phase 1: edge scatter (src row then dst row of edge_index)
    const int* src = ei;
    const int* dst = ei + E;
    ctm_edge_scatter_kernel<<<(E + 255) / 256, 256, 0, stream>>>(
        xr, xi, src, dst, ew, comm, acc, E);

    // Phase 2: WMMA expansion. 16 nodes/wave, 8 waves/block.
    int chunks = N / 16;                 // 6144
    int blocks = chunks / 8;             // 768 blocks of 256 threads
    ctm_expand_wmma_kernel<<<blocks, 256, 0, stream>>>(
        acc, Wlr, Wli, Wgr, Wgi, out);
}